// TNetkd_76158360092797
// MI455X (gfx1250) — compile-verified
//
#include <hip/hip_runtime.h>
#include <hip/hip_bf16.h>
#include <cfloat>

typedef float v2f __attribute__((ext_vector_type(2)));
typedef float v8f __attribute__((ext_vector_type(8)));

#define EPSF   1e-5f
#define SLOPEF 0.01f
#define KNN    32
#define NPTS   4096
#define NB     2

// ---- Tensor Data Mover availability (guarded; safe on host pass too) ----
#if __has_builtin(__builtin_amdgcn_tensor_load_to_lds)
#define HAVE_TDM 1
typedef unsigned int v4u __attribute__((ext_vector_type(4)));
typedef int v8i __attribute__((ext_vector_type(8)));
typedef int v4i __attribute__((ext_vector_type(4)));
#if __has_include(<hip/amd_detail/amd_gfx1250_TDM.h>)
#define TDM_6ARG 1
#endif
#endif

__device__ __forceinline__ float lrelu(float x) { return x >= 0.f ? x : SLOPEF * x; }

// ---------------------------------------------------------------- utilities
__global__ void k_zero(float* p, int n) {
    int i = blockIdx.x * blockDim.x + threadIdx.x;
    if (i < n) p[i] = 0.f;
}

// pts[b][p][c] = x[b][c][p]   (points view used for kNN + gaussian kernels)
// xe0[b][p][c] = xflat[b][p*3+c] (raw reshape view used for edge conv 1)
__global__ void k_build_pts(const float* __restrict__ x,
                            float* __restrict__ pts, float* __restrict__ xe0) {
    int i = blockIdx.x * blockDim.x + threadIdx.x;
    if (i >= NB * NPTS * 3) return;
    int c = i % 3;
    int p = (i / 3) % NPTS;
    int b = i / (3 * NPTS);
    pts[i] = x[b * 3 * NPTS + c * NPTS + p];
    xe0[i] = x[b * 3 * NPTS + p * 3 + c];
}

// ---------------------------------------------------------------- kNN, 3-D
__global__ void k_knn3(const float* __restrict__ X, int* __restrict__ idx) {
    int gid = blockIdx.x * blockDim.x + threadIdx.x;
    if (gid >= NB * NPTS) return;
    int b = gid / NPTS, q = gid % NPTS;
    const float* base = X + (size_t)b * NPTS * 3;
    float qx = base[q * 3 + 0], qy = base[q * 3 + 1], qz = base[q * 3 + 2];
    float kd[KNN]; int ki[KNN];
    for (int t = 0; t < KNN; t++) { kd[t] = FLT_MAX; ki[t] = 0; }
    for (int j = 0; j < NPTS; j++) {
        float dx = base[j * 3 + 0] - qx;
        float dy = base[j * 3 + 1] - qy;
        float dz = base[j * 3 + 2] - qz;
        float d = dx * dx + dy * dy + dz * dz;
        if (d < kd[KNN - 1]) {
            int t = KNN - 1;
            while (t > 0 && kd[t - 1] > d) { kd[t] = kd[t - 1]; ki[t] = ki[t - 1]; t--; }
            kd[t] = d; ki[t] = j;
        }
    }
    for (int t = 0; t < KNN; t++) idx[(size_t)gid * KNN + t] = ki[t];
}

// ---------------------------------------------------------------- kNN, 64-D (LDS tiled)
#define KT 128
__global__ void k_knn64(const float* __restrict__ X, int* __restrict__ idx) {
    __shared__ float tile[KT * 64];
    int b = blockIdx.y;
    int q = blockIdx.x * blockDim.x + threadIdx.x;  // blockDim = 128
    const float* base = X + (size_t)b * NPTS * 64;
    float qv[64];
#pragma unroll
    for (int c = 0; c < 64; c++) qv[c] = base[(size_t)q * 64 + c];
    float kd[KNN]; int ki[KNN];
    for (int t = 0; t < KNN; t++) { kd[t] = FLT_MAX; ki[t] = 0; }
    for (int j0 = 0; j0 < NPTS; j0 += KT) {
        __syncthreads();
        for (int t = threadIdx.x; t < KT * 64; t += blockDim.x)
            tile[t] = base[(size_t)j0 * 64 + t];
        __syncthreads();
        for (int jj = 0; jj < KT; jj++) {
            float d = 0.f;
#pragma unroll
            for (int c = 0; c < 64; c++) { float df = tile[jj * 64 + c] - qv[c]; d += df * df; }
            if (d < kd[KNN - 1]) {
                int t = KNN - 1;
                while (t > 0 && kd[t - 1] > d) { kd[t] = kd[t - 1]; ki[t] = ki[t - 1]; t--; }
                kd[t] = d; ki[t] = j0 + jj;
            }
        }
    }
    int gid = b * NPTS + q;
    for (int t = 0; t < KNN; t++) idx[(size_t)gid * KNN + t] = ki[t];
}

// ---------------------------------------------------------------- gaussian kernel conv
__global__ void k_gauss(const float* __restrict__ pts, const int* __restrict__ idx,
                        const float* __restrict__ centers, float* __restrict__ out) {
    int gid = blockIdx.x * blockDim.x + threadIdx.x;   // NB*NPTS*128
    if (gid >= NB * NPTS * 128) return;
    int o = gid & 127;
    int n = (gid >> 7) & (NPTS - 1);
    int b = gid >> 19;
    float cx = centers[o * 3 + 0], cy = centers[o * 3 + 1], cz = centers[o * 3 + 2];
    const int* id = idx + (size_t)(b * NPTS + n) * KNN;
    const float* base = pts + (size_t)b * NPTS * 3;
    float acc = 0.f;
    for (int k = 0; k < KNN; k++) {
        int j = id[k];
        float dx = base[j * 3 + 0] - cx;
        float dy = base[j * 3 + 1] - cy;
        float dz = base[j * 3 + 2] - cz;
        acc += __expf(-0.5f * (dx * dx + dy * dy + dz * dz));
    }
    out[(size_t)(b * NPTS + n) * 128 + o] = acc;
}

// ---------------------------------------------------------------- edge conv passes
template <int CIN, int O1>
__global__ void k_ec_stats1(const float* __restrict__ X, const int* __restrict__ idx,
                            const float* __restrict__ W1,
                            float* __restrict__ sum1, float* __restrict__ sq1) {
    __shared__ float ssum[O1], ssq[O1];
    for (int o = threadIdx.x; o < O1; o += blockDim.x) { ssum[o] = 0.f; ssq[o] = 0.f; }
    __syncthreads();
    int gid = blockIdx.x * blockDim.x + threadIdx.x;
    if (gid < NB * NPTS * KNN) {
        int bn = gid / KNN;
        int b = bn / NPTS;
        int j = idx[gid];
        const float* xc = X + (size_t)bn * CIN;
        const float* xn = X + (size_t)(b * NPTS + j) * CIN;
        float feat[2 * CIN];
#pragma unroll
        for (int c = 0; c < CIN; c++) { feat[c] = xn[c] - xc[c]; feat[CIN + c] = xc[c]; }
        for (int o = 0; o < O1; o++) {
            float h = 0.f;
#pragma unroll
            for (int c = 0; c < 2 * CIN; c++) h += feat[c] * W1[o * 2 * CIN + c];
            atomicAdd(&ssum[o], h);
            atomicAdd(&ssq[o], h * h);
        }
    }
    __syncthreads();
    for (int o = threadIdx.x; o < O1; o += blockDim.x) {
        atomicAdd(&sum1[o], ssum[o]);
        atomicAdd(&sq1[o], ssq[o]);
    }
}

__global__ void k_bn_finalize(const float* sum, const float* sq, float* mu, float* rs,
                              int O, float invN) {
    int o = blockIdx.x * blockDim.x + threadIdx.x;
    if (o < O) {
        float m = sum[o] * invN;
        float v = sq[o] * invN - m * m;
        mu[o] = m;
        rs[o] = rsqrtf(v + EPSF);
    }
}

template <int CIN, int O1, int O2>
__global__ void k_ec_stats2(const float* __restrict__ X, const int* __restrict__ idx,
                            const float* __restrict__ W1,
                            const float* mu1, const float* rs1,
                            const float* g1v, const float* b1v,
                            const float* __restrict__ W2,
                            float* __restrict__ sum2, float* __restrict__ sq2) {
    __shared__ float ssum[O2], ssq[O2];
    for (int o = threadIdx.x; o < O2; o += blockDim.x) { ssum[o] = 0.f; ssq[o] = 0.f; }
    __syncthreads();
    int gid = blockIdx.x * blockDim.x + threadIdx.x;
    if (gid < NB * NPTS * KNN) {
        int bn = gid / KNN;
        int b = bn / NPTS;
        int j = idx[gid];
        const float* xc = X + (size_t)bn * CIN;
        const float* xn = X + (size_t)(b * NPTS + j) * CIN;
        float feat[2 * CIN];
#pragma unroll
        for (int c = 0; c < CIN; c++) { feat[c] = xn[c] - xc[c]; feat[CIN + c] = xc[c]; }
        float a1[O1];
        for (int o = 0; o < O1; o++) {
            float h = 0.f;
#pragma unroll
            for (int c = 0; c < 2 * CIN; c++) h += feat[c] * W1[o * 2 * CIN + c];
            a1[o] = lrelu((h - mu1[o]) * rs1[o] * g1v[o] + b1v[o]);
        }
        for (int o = 0; o < O2; o++) {
            float h = 0.f;
#pragma unroll
            for (int c = 0; c < O1; c++) h += a1[c] * W2[o * O1 + c];
            atomicAdd(&ssum[o], h);
            atomicAdd(&ssq[o], h * h);
        }
    }
    __syncthreads();
    for (int o = threadIdx.x; o < O2; o += blockDim.x) {
        atomicAdd(&sum2[o], ssum[o]);
        atomicAdd(&sq2[o], ssq[o]);
    }
}

template <int CIN, int O1, int O2>
__global__ void k_ec_out(const float* __restrict__ X, const int* __restrict__ idx,
                         const float* __restrict__ W1,
                         const float* mu1, const float* rs1,
                         const float* g1v, const float* b1v,
                         const float* __restrict__ W2,
                         const float* mu2, const float* rs2,
                         const float* g2v, const float* b2v,
                         float* __restrict__ out) {
    int gid = blockIdx.x * blockDim.x + threadIdx.x;   // NB*NPTS
    if (gid >= NB * NPTS) return;
    int b = gid / NPTS;
    const float* xc = X + (size_t)gid * CIN;
    const int* id = idx + (size_t)gid * KNN;
    float mx[O2];
#pragma unroll
    for (int o = 0; o < O2; o++) mx[o] = -FLT_MAX;
    for (int k = 0; k < KNN; k++) {
        int j = id[k];
        const float* xn = X + (size_t)(b * NPTS + j) * CIN;
        float feat[2 * CIN];
#pragma unroll
        for (int c = 0; c < CIN; c++) { feat[c] = xn[c] - xc[c]; feat[CIN + c] = xc[c]; }
        float a1[O1];
        for (int o = 0; o < O1; o++) {
            float h = 0.f;
#pragma unroll
            for (int c = 0; c < 2 * CIN; c++) h += feat[c] * W1[o * 2 * CIN + c];
            a1[o] = lrelu((h - mu1[o]) * rs1[o] * g1v[o] + b1v[o]);
        }
        for (int o = 0; o < O2; o++) {
            float h = 0.f;
#pragma unroll
            for (int c = 0; c < O1; c++) h += a1[c] * W2[o * O1 + c];
            float a2 = lrelu((h - mu2[o]) * rs2[o] * g2v[o] + b2v[o]);
            mx[o] = fmaxf(mx[o], a2);
        }
    }
    for (int o = 0; o < O2; o++) out[(size_t)gid * O2 + o] = mx[o];
}

// ---------------------------------------------------------------- concat -> (2,256,4096)
__global__ void k_hcat(const float* __restrict__ xe2, const float* __restrict__ gk,
                       float* __restrict__ hcat) {
    int gid = blockIdx.x * blockDim.x + threadIdx.x;   // NB*256*NPTS
    if (gid >= NB * 256 * NPTS) return;
    int p = gid & (NPTS - 1);
    int c = (gid >> 12) & 255;
    int b = gid >> 20;
    float v;
    if (c < 128)
        v = xe2[(size_t)b * 128 * NPTS + (size_t)c * NPTS + p];  // raw reshape of (n,128)
    else
        v = gk[(size_t)(b * NPTS + p) * 128 + (c - 128)];        // transpose of kernels
    hcat[gid] = v;
}

// ---------------------------------------------------------------- fp32 WMMA GEMM
// Y[b][m][n] = sum_k W[m][k] * H[b][k][n], N = 4096.
// Block = 128 threads (4 waves) computing a 64(M) x 64(N) tile.
// B K-slab (16 x 64) staged in LDS (via TDM when available) and shared by all
// 4 waves; each wave holds 4 accumulators (16x64) and reuses its A fragment
// across 4 wmmas -> 16 v_wmma_f32_16x16x4_f32 per 4KB staged tile.
#define GEMM_KC 16
__global__ void k_gemm_wmma(const float* __restrict__ W, const float* __restrict__ H,
                            float* __restrict__ Y, int M, int K) {
    const int N = NPTS;
    __shared__ float sB[GEMM_KC * 64];
    int lane = threadIdx.x & 31;
    int wv = threadIdx.x >> 5;             // 0..3 -> which 16-row strip
    int lm = lane & 15;
    int lh = lane >> 4;                    // 0 or 1
    int n0 = blockIdx.x * 64;
    int m0 = blockIdx.y * 64 + wv * 16;
    int b = blockIdx.z;
    const float* Hb = H + (size_t)b * K * N;

    v8f acc0 = {0.f, 0.f, 0.f, 0.f, 0.f, 0.f, 0.f, 0.f};
    v8f acc1 = acc0, acc2 = acc0, acc3 = acc0;

    for (int k0 = 0; k0 < K; k0 += GEMM_KC) {
        __syncthreads();   // everyone done reading previous slab
#ifdef HAVE_TDM
        if (threadIdx.x < 32) {  // wave 0 drives the Tensor Data Mover
            const float* gsrc = Hb + (size_t)k0 * N + n0;
            unsigned long long ga = (unsigned long long)(uintptr_t)gsrc;
            unsigned int lds = (unsigned int)(uintptr_t)(void*)sB;
            // D# group0: count=1 | lds_addr | global_addr[31:0] | addr[56:32]+type=2
            v4u g0 = {1u, lds, (unsigned int)ga,
                      (unsigned int)(((ga >> 32) & 0x1FFFFFFu) | (2u << 30))};
            // D# group1: data_size=4B; tensor_dim0=N, tensor_dim1=K;
            // tile_dim0=64, tile_dim1=GEMM_KC; dim0_stride=N
            v8i g1 = {(int)(2u << 16),
                      (int)((unsigned)(N & 0xFFFF) << 16),
                      (int)(((unsigned)N >> 16) | ((unsigned)(K & 0xFFFF) << 16)),
                      (int)(((unsigned)K >> 16) | (64u << 16)),
                      (int)GEMM_KC,
                      (int)N,
                      0, 0};
            v4i g2 = {0, 0, 0, 0};
            v4i g3 = {0, 0, 0, 0};
#ifdef TDM_6ARG
            v8i g4 = {0, 0, 0, 0, 0, 0, 0, 0};
            __builtin_amdgcn_tensor_load_to_lds(g0, g1, g2, g3, g4, 0);
#else
            __builtin_amdgcn_tensor_load_to_lds(g0, g1, g2, g3, 0);
#endif
            __builtin_amdgcn_s_wait_tensorcnt(0);
        }
#else
        for (int t = threadIdx.x; t < GEMM_KC * 64; t += blockDim.x)
            sB[t] = Hb[(size_t)(k0 + (t >> 6)) * N + n0 + (t & 63)];
#endif
        __syncthreads();   // slab visible to all waves
#pragma unroll
        for (int kk = 0; kk < GEMM_KC; kk += 4) {
            // A 16x4 fragment: lane(M=lm), reg v holds K = k0+kk + v + 2*lh
            const float* wrow = W + (size_t)(m0 + lm) * K + k0 + kk + 2 * lh;
            v2f a; a[0] = wrow[0]; a[1] = wrow[1];
            // B 4x16 fragments from LDS: reg v holds K-row kk+v+2*lh, col nt*16+lm
            const float* srow = &sB[(kk + 2 * lh) * 64];
            v2f b0; b0[0] = srow[lm +  0]; b0[1] = srow[64 + lm +  0];
            v2f b1; b1[0] = srow[lm + 16]; b1[1] = srow[64 + lm + 16];
            v2f b2; b2[0] = srow[lm + 32]; b2[1] = srow[64 + lm + 32];
            v2f b3; b3[0] = srow[lm + 48]; b3[1] = srow[64 + lm + 48];
            acc0 = __builtin_amdgcn_wmma_f32_16x16x4_f32(false, a, false, b0, (short)0, acc0, false, false);
            acc1 = __builtin_amdgcn_wmma_f32_16x16x4_f32(false, a, false, b1, (short)0, acc1, false, false);
            acc2 = __builtin_amdgcn_wmma_f32_16x16x4_f32(false, a, false, b2, (short)0, acc2, false, false);
            acc3 = __builtin_amdgcn_wmma_f32_16x16x4_f32(false, a, false, b3, (short)0, acc3, false, false);
        }
    }
    // D 16x16: reg v -> row m0 + v + 8*lh, col n0 + nt*16 + lm
    float* yb = Y + (size_t)b * M * N + (size_t)(m0 + 8 * lh) * N + n0 + lm;
#pragma unroll
    for (int v = 0; v < 8; v++) {
        yb[(size_t)v * N +  0] = acc0[v];
        yb[(size_t)v * N + 16] = acc1[v];
        yb[(size_t)v * N + 32] = acc2[v];
        yb[(size_t)v * N + 48] = acc3[v];
    }
}

// ---------------------------------------------------------------- per-channel BN stats
__global__ void k_chan_stats(const float* __restrict__ Y, float* mu, float* rs, int M) {
    const int N = NPTS;
    int o = blockIdx.x;
    float s = 0.f, sq = 0.f;
    for (int i = threadIdx.x; i < NB * N; i += blockDim.x) {
        int b = i >> 12;
        int p = i & (N - 1);
        float v = Y[(size_t)b * M * N + (size_t)o * N + p];
        s += v; sq += v * v;
    }
    __shared__ float rs_s[256], rq_s[256];
    rs_s[threadIdx.x] = s; rq_s[threadIdx.x] = sq;
    __syncthreads();
    for (int st = 128; st > 0; st >>= 1) {
        if (threadIdx.x < st) {
            rs_s[threadIdx.x] += rs_s[threadIdx.x + st];
            rq_s[threadIdx.x] += rq_s[threadIdx.x + st];
        }
        __syncthreads();
    }
    if (threadIdx.x == 0) {
        float invN = 1.f / (float)(NB * N);
        float m = rs_s[0] * invN;
        float var = rq_s[0] * invN - m * m;
        mu[o] = m;
        rs[o] = rsqrtf(var + EPSF);
    }
}

__global__ void k_bn_act(float* __restrict__ Y, const float* g, const float* bta,
                         const float* mu, const float* rs, int M) {
    int gid = blockIdx.x * blockDim.x + threadIdx.x;
    if (gid >= NB * M * NPTS) return;
    int o = (gid >> 12) % M;
    float v = Y[gid];
    v = (v - mu[o]) * rs[o] * g[o] + bta[o];
    Y[gid] = lrelu(v);
}

// ---------------------------------------------------------------- final head
__global__ void k_final(const float* __restrict__ H3, const float* __restrict__ W4,
                        float* __restrict__ out) {
    const int N = NPTS;
    int b = blockIdx.x / 9, o = blockIdx.x % 9;
    __shared__ float w[256];
    for (int c = threadIdx.x; c < 256; c += blockDim.x) w[c] = W4[o * 256 + c];
    __syncthreads();
    float m = -FLT_MAX;
    for (int p = threadIdx.x; p < N; p += blockDim.x) {
        float acc = 0.f;
#pragma unroll 8
        for (int c = 0; c < 256; c++)
            acc += H3[(size_t)b * 256 * N + (size_t)c * N + p] * w[c];
        m = fmaxf(m, acc);
    }
    __shared__ float red[256];
    red[threadIdx.x] = m;
    __syncthreads();
    for (int st = 128; st > 0; st >>= 1) {
        if (threadIdx.x < st) red[threadIdx.x] = fmaxf(red[threadIdx.x], red[threadIdx.x + st]);
        __syncthreads();
    }
    if (threadIdx.x == 0) {
        float e = (o == 0 || o == 4 || o == 8) ? 1.f : 0.f;
        out[b * 9 + o] = red[0] + e;
    }
}

// ---------------------------------------------------------------- launcher
extern "C" void kernel_launch(void* const* d_in, const int* in_sizes, int n_in,
                              void* d_out, int out_size, void* d_ws, size_t ws_size,
                              hipStream_t stream) {
    (void)in_sizes; (void)n_in; (void)out_size; (void)ws_size;
    const float* x     = (const float*)d_in[0];
    const float* kc    = (const float*)d_in[1];
    const float* e1_w1 = (const float*)d_in[2];
    const float* e1_g1 = (const float*)d_in[3];
    const float* e1_b1 = (const float*)d_in[4];
    const float* e1_w2 = (const float*)d_in[5];
    const float* e1_g2 = (const float*)d_in[6];
    const float* e1_b2 = (const float*)d_in[7];
    const float* e2_w1 = (const float*)d_in[8];
    const float* e2_g1 = (const float*)d_in[9];
    const float* e2_b1 = (const float*)d_in[10];
    const float* e2_w2 = (const float*)d_in[11];
    const float* e2_g2 = (const float*)d_in[12];
    const float* e2_b2 = (const float*)d_in[13];
    const float* w1 = (const float*)d_in[14];
    const float* g1 = (const float*)d_in[15];
    const float* b1 = (const float*)d_in[16];
    const float* w2 = (const float*)d_in[17];
    const float* g2 = (const float*)d_in[18];
    const float* b2 = (const float*)d_in[19];
    const float* w3 = (const float*)d_in[20];
    const float* g3 = (const float*)d_in[21];
    const float* b3 = (const float*)d_in[22];
    const float* w4 = (const float*)d_in[23];

    float* ws = (float*)d_ws;
    // workspace layout (floats)
    const size_t OFF_PTS = 0;                    // 24576
    const size_t OFF_XE0 = 24576;                // 24576
    const size_t OFF_IDXA = 49152;               // 262144 (int)
    const size_t OFF_IDXB = 311296;              // 262144 (int)
    const size_t OFF_IDXC = 573440;              // 262144 (int)
    const size_t OFF_GK = 835584;                // 1048576
    const size_t OFF_XE1 = 1884160;              // 524288
    const size_t OFF_XE2 = 2408448;              // 1048576
    const size_t OFF_STATS = 3457024;            // 8192
    const size_t OFF_HCAT = 3465216;             // 2097152
    const size_t OFF_Y1 = 5562368;               // 8388608
    const size_t OFF_Y2 = 13950976;              // 4194304
    const size_t OFF_Y3 = OFF_HCAT;              // reuse (hcat dead after y1 gemm)

    float* pts = ws + OFF_PTS;
    float* xe0 = ws + OFF_XE0;
    int* idxA = (int*)(ws + OFF_IDXA);
    int* idxB = (int*)(ws + OFF_IDXB);
    int* idxC = (int*)(ws + OFF_IDXC);
    float* gk = ws + OFF_GK;
    float* xe1 = ws + OFF_XE1;
    float* xe2 = ws + OFF_XE2;
    float* S = ws + OFF_STATS;
    float* hcat = ws + OFF_HCAT;
    float* y1 = ws + OFF_Y1;
    float* y2 = ws + OFF_Y2;
    float* y3 = ws + OFF_Y3;

    // stats sub-offsets
    float* e1s1 = S + 0;    float* e1q1 = S + 64;
    float* e1m1 = S + 128;  float* e1r1 = S + 192;
    float* e1s2 = S + 256;  float* e1q2 = S + 320;
    float* e1m2 = S + 384;  float* e1r2 = S + 448;
    float* e2s1 = S + 512;  float* e2q1 = S + 640;
    float* e2m1 = S + 768;  float* e2r1 = S + 896;
    float* e2s2 = S + 1024; float* e2q2 = S + 1152;
    float* e2m2 = S + 1280; float* e2r2 = S + 1408;
    float* c1mu = S + 1536; float* c1rs = S + 2560;
    float* c2mu = S + 3584; float* c2rs = S + 4096;
    float* c3mu = S + 4608; float* c3rs = S + 4864;

    const float invEC = 1.f / (float)(NB * NPTS * KNN);

    k_zero<<<(1536 + 255) / 256, 256, 0, stream>>>(S, 1536);
    k_build_pts<<<(NB * NPTS * 3 + 255) / 256, 256, 0, stream>>>(x, pts, xe0);
    k_knn3<<<(NB * NPTS + 63) / 64, 64, 0, stream>>>(pts, idxA);
    k_knn3<<<(NB * NPTS + 63) / 64, 64, 0, stream>>>(xe0, idxB);
    k_gauss<<<(NB * NPTS * 128) / 256, 256, 0, stream>>>(pts, idxA, kc, gk);

    // edge conv 1 (3 -> 64 -> 64)
    k_ec_stats1<3, 64><<<(NB * NPTS * KNN + 255) / 256, 256, 0, stream>>>(xe0, idxB, e1_w1, e1s1, e1q1);
    k_bn_finalize<<<1, 64, 0, stream>>>(e1s1, e1q1, e1m1, e1r1, 64, invEC);
    k_ec_stats2<3, 64, 64><<<(NB * NPTS * KNN + 255) / 256, 256, 0, stream>>>(
        xe0, idxB, e1_w1, e1m1, e1r1, e1_g1, e1_b1, e1_w2, e1s2, e1q2);
    k_bn_finalize<<<1, 64, 0, stream>>>(e1s2, e1q2, e1m2, e1r2, 64, invEC);
    k_ec_out<3, 64, 64><<<(NB * NPTS + 63) / 64, 64, 0, stream>>>(
        xe0, idxB, e1_w1, e1m1, e1r1, e1_g1, e1_b1, e1_w2, e1m2, e1r2, e1_g2, e1_b2, xe1);

    k_knn64<<<dim3(NPTS / KT, NB), KT, 0, stream>>>(xe1, idxC);

    // edge conv 2 (64 -> 128 -> 128)
    k_ec_stats1<64, 128><<<(NB * NPTS * KNN + 255) / 256, 256, 0, stream>>>(xe1, idxC, e2_w1, e2s1, e2q1);
    k_bn_finalize<<<1, 128, 0, stream>>>(e2s1, e2q1, e2m1, e2r1, 128, invEC);
    k_ec_stats2<64, 128, 128><<<(NB * NPTS * KNN + 255) / 256, 256, 0, stream>>>(
        xe1, idxC, e2_w1, e2m1, e2r1, e2_g1, e2_b1, e2_w2, e2s2, e2q2);
    k_bn_finalize<<<1, 128, 0, stream>>>(e2s2, e2q2, e2m2, e2r2, 128, invEC);
    k_ec_out<64, 128, 128><<<(NB * NPTS + 63) / 64, 64, 0, stream>>>(
        xe1, idxC, e2_w1, e2m1, e2r1, e2_g1, e2_b1, e2_w2, e2m2, e2r2, e2_g2, e2_b2, xe2);

    k_hcat<<<(NB * 256 * NPTS) / 256, 256, 0, stream>>>(xe2, gk, hcat);

    // cbr1: 1024 x 256
    k_gemm_wmma<<<dim3(NPTS / 64, 1024 / 64, NB), 128, 0, stream>>>(w1, hcat, y1, 1024, 256);
    k_chan_stats<<<1024, 256, 0, stream>>>(y1, c1mu, c1rs, 1024);
    k_bn_act<<<(NB * 1024 * NPTS) / 256, 256, 0, stream>>>(y1, g1, b1, c1mu, c1rs, 1024);
    // cbr2: 512 x 1024
    k_gemm_wmma<<<dim3(NPTS / 64, 512 / 64, NB), 128, 0, stream>>>(w2, y1, y2, 512, 1024);
    k_chan_stats<<<512, 256, 0, stream>>>(y2, c2mu, c2rs, 512);
    k_bn_act<<<(NB * 512 * NPTS) / 256, 256, 0, stream>>>(y2, g2, b2, c2mu, c2rs, 512);
    // cbr3: 256 x 512
    k_gemm_wmma<<<dim3(NPTS / 64, 256 / 64, NB), 128, 0, stream>>>(w3, y2, y3, 256, 512);
    k_chan_stats<<<256, 256, 0, stream>>>(y3, c3mu, c3rs, 256);
    k_bn_act<<<(NB * 256 * NPTS) / 256, 256, 0, stream>>>(y3, g3, b3, c3mu, c3rs, 256);

    k_final<<<NB * 9, 256, 0, stream>>>(y3, w4, (float*)d_out);
}